// GAT_62225486184529
// MI455X (gfx1250) — compile-verified
//
#include <hip/hip_runtime.h>
#include <math.h>

#define N_NODES 50000
#define N_EDGES 800000
#define IN_F    191
#define H1 16
#define C1 16
#define F1 256   /* H1*C1 */
#define H2 8
#define C2 3
#define F2 24    /* H2*C2 */
#define E_TOT (N_EDGES + N_NODES)

typedef __attribute__((ext_vector_type(2))) float v2f;
typedef __attribute__((ext_vector_type(8))) float v8f;

// ---------- helpers ----------

// Order-preserving float<->uint encoding for atomicMax-based segment max.
__device__ __forceinline__ unsigned enc_f32(float f) {
  unsigned u = __float_as_uint(f);
  return (u & 0x80000000u) ? ~u : (u | 0x80000000u);
}
__device__ __forceinline__ float dec_f32(unsigned u) {
  return (u & 0x80000000u) ? __uint_as_float(u & 0x7fffffffu)
                           : __uint_as_float(~u);
}
#define MX_INIT 0x00800000u  /* enc(-FLT_MAX) */

__device__ __forceinline__ void atomic_add_f32(float* p, float v) {
  __hip_atomic_fetch_add(p, v, __ATOMIC_RELAXED, __HIP_MEMORY_SCOPE_AGENT);
}

// ---------- prep: mean(edge_attr), ce coefficients ----------
__global__ void __launch_bounds__(1024)
prep_kernel(const float* __restrict__ ea,
            const float* __restrict__ We1, const float* __restrict__ ae1,
            const float* __restrict__ We2, const float* __restrict__ ae2,
            float* __restrict__ mean_out, float* __restrict__ ce1,
            float* __restrict__ ce2) {
  __shared__ float red[1024];
  const int t = threadIdx.x;
  float s = 0.f;
  for (int i = t; i < N_EDGES; i += 1024) s += ea[i];
  red[t] = s;
  __syncthreads();
  for (int off = 512; off > 0; off >>= 1) {
    if (t < off) red[t] += red[t + off];
    __syncthreads();
  }
  if (t == 0) mean_out[0] = red[0] * (1.0f / (float)N_EDGES);
  if (t < H1) {
    float c = 0.f;
    for (int k = 0; k < C1; k++) c += We1[t * C1 + k] * ae1[t * C1 + k];
    ce1[t] = c;
  } else if (t >= 32 && t < 32 + H2) {
    const int h = t - 32;
    float c = 0.f;
    for (int k = 0; k < C2; k++) c += We2[h * C2 + k] * ae2[h * C2 + k];
    ce2[h] = c;
  }
}

// ---------- init segment-max keys ----------
__global__ void init_keys_kernel(unsigned* __restrict__ mx1,
                                 unsigned* __restrict__ mx2) {
  const int idx = blockIdx.x * blockDim.x + threadIdx.x;
  if (idx < N_NODES * H1) mx1[idx] = MX_INIT;
  if (idx < N_NODES * H2) mx2[idx] = MX_INIT;
}

// ---------- GEMM1: xl1[50000,256] = x[50000,191] @ W1[191,256] (fp32 WMMA) ----------
// Block: 256 threads = 8 waves; block tile 16(M) x 128(N); grid (3125, 2).
// A-frag ISA layout: lanes 0-15 -> K={k,k+1}, lanes 16-31 -> K={k+2,k+3}, M=lane&15.
// B-frag mirrored; D: VGPR i -> M = i + 8*(lane>>4), N = lane&15.
__global__ void __launch_bounds__(256)
gemm1_wmma(const float* __restrict__ x, const float* __restrict__ W1,
           float* __restrict__ xl1) {
  __shared__ float As[16][49];    // pad 49 (odd): conflict-free column reads
  __shared__ float Bs[48][128];
  const int row0 = blockIdx.x * 16;
  const int colB = blockIdx.y * 128;
  const int t = threadIdx.x;
  const int wave = t >> 5, lane = t & 31, lh = lane & 15, hi = lane >> 4;
  v8f acc = {};
  for (int k0 = 0; k0 < 192; k0 += 48) {
    // A tile: x rows are 191 floats (misaligned) -> scalar dword loads.
    for (int idx = t; idx < 16 * 48; idx += 256) {
      const int r = idx / 48, c = idx % 48, k = k0 + c;
      As[r][c] = (k < IN_F) ? x[(row0 + r) * IN_F + k] : 0.f;
    }
    // B tile: W1 rows are 256 floats, 16B aligned -> float4 loads/stores.
    for (int idx4 = t; idx4 < (48 * 128) / 4; idx4 += 256) {
      const int r = idx4 >> 5, c = (idx4 & 31) << 2, k = k0 + r;
      float4 f = make_float4(0.f, 0.f, 0.f, 0.f);
      if (k < IN_F) {
        f = *(const float4*)&W1[k * F1 + colB + c];
        // sparse cacheline-granular prefetch of next K-chunk
        if ((idx4 & 7) == 0 && k + 48 < IN_F)
          __builtin_prefetch(&W1[(k + 48) * F1 + colB + c], 0, 0);
      }
      *(float4*)&Bs[r][c] = f;
    }
    __syncthreads();
#pragma unroll
    for (int kk = 0; kk < 48; kk += 4) {
      v2f a, b;
      a.x = As[lh][kk + 2 * hi];
      a.y = As[lh][kk + 2 * hi + 1];
      b.x = Bs[kk + 2 * hi][wave * 16 + lh];
      b.y = Bs[kk + 2 * hi + 1][wave * 16 + lh];
      acc = __builtin_amdgcn_wmma_f32_16x16x4_f32(false, a, false, b,
                                                  (short)0, acc, false, false);
    }
    __syncthreads();
  }
  const int col = colB + wave * 16 + lh;
#pragma unroll
  for (int i = 0; i < 8; i++)
    xl1[(row0 + i + 8 * hi) * F1 + col] = acc[i];
}

// ---------- GEMM2: xl2[50000,24] = h1[50000,256] @ W2[256,24] (fp32 WMMA) ----------
// Block: 64 threads = 2 waves; tile 16(M) x 32(N) (cols >=24 masked). Grid 3125.
__global__ void __launch_bounds__(64)
gemm2_wmma(const float* __restrict__ h1, const float* __restrict__ W2,
           float* __restrict__ xl2) {
  __shared__ float As[16][258];   // stride 258: even-spread banks for column reads
  const int row0 = blockIdx.x * 16;
  const int t = threadIdx.x;
  const int wave = t >> 5, lane = t & 31, lh = lane & 15, hi = lane >> 4;
  // h1 rows are 256 floats, 16B aligned -> float4 staging.
  for (int idx4 = t; idx4 < (16 * 256) / 4; idx4 += 64) {
    const int r = idx4 >> 6, c = (idx4 & 63) << 2;
    const float4 f = *(const float4*)&h1[(row0 + r) * F1 + c];
    As[r][c + 0] = f.x; As[r][c + 1] = f.y;
    As[r][c + 2] = f.z; As[r][c + 3] = f.w;
  }
  __syncthreads();
  const int col = wave * 16 + lh;
  v8f acc = {};
  for (int k = 0; k < 256; k += 4) {
    v2f a, b;
    a.x = As[lh][k + 2 * hi];
    a.y = As[lh][k + 2 * hi + 1];
    if (col < F2) {
      b.x = W2[(k + 2 * hi) * F2 + col];
      b.y = W2[(k + 2 * hi + 1) * F2 + col];
    } else {
      b.x = 0.f; b.y = 0.f;
    }
    acc = __builtin_amdgcn_wmma_f32_16x16x4_f32(false, a, false, b,
                                                (short)0, acc, false, false);
  }
  if (col < F2) {
#pragma unroll
    for (int i = 0; i < 8; i++)
      xl2[(row0 + i + 8 * hi) * F2 + col] = acc[i];
  }
}

// ---------- per-node attention logits ----------
template <int H, int C>
__global__ void alpha_node_kernel(const float* __restrict__ xl,
                                  const float* __restrict__ a_src,
                                  const float* __restrict__ a_dst,
                                  float* __restrict__ asrc,
                                  float* __restrict__ adst) {
  const int idx = blockIdx.x * blockDim.x + threadIdx.x;
  if (idx >= N_NODES * H) return;
  const int n = idx / H, h = idx % H;
  const float* row = xl + n * (H * C) + h * C;
  float s = 0.f, d = 0.f;
#pragma unroll
  for (int c = 0; c < C; c++) {
    const float v = row[c];
    s += v * a_src[h * C + c];
    d += v * a_dst[h * C + c];
  }
  asrc[idx] = s;
  adst[idx] = d;
}

// ---------- per-edge leaky-relu logit (shared by the 3 softmax passes) ----------
template <int H>
__device__ __forceinline__ float edge_logit(int e, int h, const int* __restrict__ ei,
                                            const float* __restrict__ ea,
                                            const float* __restrict__ mean,
                                            const float* __restrict__ ce,
                                            const float* __restrict__ asrc,
                                            const float* __restrict__ adst,
                                            int& s, int& d) {
  float eav;
  if (e < N_EDGES) {
    s = ei[e];
    d = ei[N_EDGES + e];
    eav = ea[e];
  } else {                       // appended self loops, edge_attr = mean
    s = d = e - N_EDGES;
    eav = mean[0];
  }
  const float a = asrc[s * H + h] + adst[d * H + h] + eav * ce[h];
  return (a > 0.f) ? a : 0.2f * a;   // leaky_relu(0.2)
}

template <int H>
__global__ void edge_max_kernel(const int* __restrict__ ei,
                                const float* __restrict__ ea,
                                const float* __restrict__ mean,
                                const float* __restrict__ ce,
                                const float* __restrict__ asrc,
                                const float* __restrict__ adst,
                                unsigned* __restrict__ mx) {
  const int idx = blockIdx.x * blockDim.x + threadIdx.x;
  if (idx >= E_TOT * H) return;
  int s, d;
  const int e = idx / H, h = idx % H;
  const float a = edge_logit<H>(e, h, ei, ea, mean, ce, asrc, adst, s, d);
  atomicMax(&mx[d * H + h], enc_f32(a));
}

template <int H>
__global__ void edge_den_kernel(const int* __restrict__ ei,
                                const float* __restrict__ ea,
                                const float* __restrict__ mean,
                                const float* __restrict__ ce,
                                const float* __restrict__ asrc,
                                const float* __restrict__ adst,
                                const unsigned* __restrict__ mx,
                                float* __restrict__ den) {
  const int idx = blockIdx.x * blockDim.x + threadIdx.x;
  if (idx >= E_TOT * H) return;
  int s, d;
  const int e = idx / H, h = idx % H;
  const float a = edge_logit<H>(e, h, ei, ea, mean, ce, asrc, adst, s, d);
  const float m = dec_f32(mx[d * H + h]);
  atomic_add_f32(&den[d * H + h], __expf(a - m));
}

// Layer-1 aggregation: C1=16 contiguous floats per (node,head), 64B aligned
// -> float4 gathers (L2 resident) + scalar fp32 atomics into agg.
__global__ void edge_agg1_kernel(const int* __restrict__ ei,
                                 const float* __restrict__ ea,
                                 const float* __restrict__ mean,
                                 const float* __restrict__ ce,
                                 const float* __restrict__ asrc,
                                 const float* __restrict__ adst,
                                 const unsigned* __restrict__ mx,
                                 const float* __restrict__ den,
                                 const float* __restrict__ xl,
                                 float* __restrict__ agg) {
  const int idx = blockIdx.x * blockDim.x + threadIdx.x;
  if (idx >= E_TOT * H1) return;
  int s, d;
  const int e = idx / H1, h = idx % H1;
  const float a = edge_logit<H1>(e, h, ei, ea, mean, ce, asrc, adst, s, d);
  const float4* srow = (const float4*)(xl + s * F1 + h * C1);
  __builtin_prefetch(srow, 0, 0);   // start the gather before exp/div resolves
  const float m = dec_f32(mx[d * H1 + h]);
  const float w = __expf(a - m) / (den[d * H1 + h] + 1e-16f);
  float* drow = agg + d * F1 + h * C1;
#pragma unroll
  for (int q = 0; q < C1 / 4; q++) {
    const float4 v = srow[q];
    atomic_add_f32(&drow[4 * q + 0], w * v.x);
    atomic_add_f32(&drow[4 * q + 1], w * v.y);
    atomic_add_f32(&drow[4 * q + 2], w * v.z);
    atomic_add_f32(&drow[4 * q + 3], w * v.w);
  }
}

// Layer-2 aggregation: C2=3 -> scalar.
__global__ void edge_agg2_kernel(const int* __restrict__ ei,
                                 const float* __restrict__ ea,
                                 const float* __restrict__ mean,
                                 const float* __restrict__ ce,
                                 const float* __restrict__ asrc,
                                 const float* __restrict__ adst,
                                 const unsigned* __restrict__ mx,
                                 const float* __restrict__ den,
                                 const float* __restrict__ xl,
                                 float* __restrict__ agg) {
  const int idx = blockIdx.x * blockDim.x + threadIdx.x;
  if (idx >= E_TOT * H2) return;
  int s, d;
  const int e = idx / H2, h = idx % H2;
  const float a = edge_logit<H2>(e, h, ei, ea, mean, ce, asrc, adst, s, d);
  const float* srow = xl + s * F2 + h * C2;
  __builtin_prefetch(srow, 0, 0);
  const float m = dec_f32(mx[d * H2 + h]);
  const float w = __expf(a - m) / (den[d * H2 + h] + 1e-16f);
  float* drow = agg + d * F2 + h * C2;
#pragma unroll
  for (int c = 0; c < C2; c++) atomic_add_f32(&drow[c], w * srow[c]);
}

// ---------- h1 = elu(agg1 + b1), in place, float4 ----------
__global__ void finalize1_kernel(float* __restrict__ agg1,
                                 const float* __restrict__ b1) {
  const int idx4 = blockIdx.x * blockDim.x + threadIdx.x;
  if (idx4 >= (N_NODES * F1) / 4) return;
  const int e0 = idx4 << 2;
  float4 v = *(float4*)&agg1[e0];
  const float4 b = *(const float4*)&b1[e0 & (F1 - 1)];
  v.x += b.x; v.y += b.y; v.z += b.z; v.w += b.w;
  v.x = (v.x > 0.f) ? v.x : expm1f(v.x);
  v.y = (v.y > 0.f) ? v.y : expm1f(v.y);
  v.z = (v.z > 0.f) ? v.z : expm1f(v.z);
  v.w = (v.w > 0.f) ? v.w : expm1f(v.w);
  *(float4*)&agg1[e0] = v;
}

// ---------- out = elu(mean_h(agg2) + b2) ----------
__global__ void final2_kernel(const float* __restrict__ agg2,
                              const float* __restrict__ b2,
                              float* __restrict__ out) {
  const int idx = blockIdx.x * blockDim.x + threadIdx.x;
  if (idx >= N_NODES * C2) return;
  const int n = idx / C2, c = idx % C2;
  float s = 0.f;
#pragma unroll
  for (int h = 0; h < H2; h++) s += agg2[n * F2 + h * C2 + c];
  const float v = s * (1.f / (float)H2) + b2[c];
  out[idx] = (v > 0.f) ? v : expm1f(v);
}

// ---------- host ----------
extern "C" void kernel_launch(void* const* d_in, const int* in_sizes, int n_in,
                              void* d_out, int out_size, void* d_ws, size_t ws_size,
                              hipStream_t stream) {
  (void)in_sizes; (void)n_in; (void)out_size; (void)ws_size;
  const float* x   = (const float*)d_in[0];
  const int*   ei  = (const int*)d_in[1];
  const float* ea  = (const float*)d_in[2];
  const float* W1  = (const float*)d_in[3];
  const float* We1 = (const float*)d_in[4];
  const float* as1 = (const float*)d_in[5];
  const float* ad1 = (const float*)d_in[6];
  const float* ae1 = (const float*)d_in[7];
  const float* b1  = (const float*)d_in[8];
  const float* W2  = (const float*)d_in[9];
  const float* We2 = (const float*)d_in[10];
  const float* as2 = (const float*)d_in[11];
  const float* ad2 = (const float*)d_in[12];
  const float* ae2 = (const float*)d_in[13];
  const float* b2  = (const float*)d_in[14];
  float* out = (float*)d_out;

  // workspace layout (floats)
  float* w = (float*)d_ws;
  float* mean = w;              // 1
  float* ce1  = w + 1;          // 16
  float* ce2  = w + 17;         // 8
  size_t off = 32;
  float* xl1   = w + off; off += (size_t)N_NODES * F1;
  float* asrc1 = w + off; off += (size_t)N_NODES * H1;
  float* adst1 = w + off; off += (size_t)N_NODES * H1;
  unsigned* mx1 = (unsigned*)(w + off); off += (size_t)N_NODES * H1;
  float* den1  = w + off; off += (size_t)N_NODES * H1;
  float* agg1  = w + off; off += (size_t)N_NODES * F1;   // becomes h1 in place
  float* xl2   = w + off; off += (size_t)N_NODES * F2;
  float* asrc2 = w + off; off += (size_t)N_NODES * H2;
  float* adst2 = w + off; off += (size_t)N_NODES * H2;
  unsigned* mx2 = (unsigned*)(w + off); off += (size_t)N_NODES * H2;
  float* den2  = w + off; off += (size_t)N_NODES * H2;
  float* agg2  = w + off; off += (size_t)N_NODES * F2;

  // accumulator init (graph-capture-safe)
  hipMemsetAsync(den1, 0, (size_t)N_NODES * H1 * 4, stream);
  hipMemsetAsync(agg1, 0, (size_t)N_NODES * F1 * 4, stream);
  hipMemsetAsync(den2, 0, (size_t)N_NODES * H2 * 4, stream);
  hipMemsetAsync(agg2, 0, (size_t)N_NODES * F2 * 4, stream);
  init_keys_kernel<<<(N_NODES * H1 + 255) / 256, 256, 0, stream>>>(mx1, mx2);

  prep_kernel<<<1, 1024, 0, stream>>>(ea, We1, ae1, We2, ae2, mean, ce1, ce2);

  // ----- layer 1 -----
  gemm1_wmma<<<dim3(N_NODES / 16, 2), 256, 0, stream>>>(x, W1, xl1);
  alpha_node_kernel<H1, C1><<<(N_NODES * H1 + 255) / 256, 256, 0, stream>>>(
      xl1, as1, ad1, asrc1, adst1);
  const int g1 = (E_TOT * H1 + 255) / 256;
  edge_max_kernel<H1><<<g1, 256, 0, stream>>>(ei, ea, mean, ce1, asrc1, adst1, mx1);
  edge_den_kernel<H1><<<g1, 256, 0, stream>>>(ei, ea, mean, ce1, asrc1, adst1, mx1, den1);
  edge_agg1_kernel<<<g1, 256, 0, stream>>>(ei, ea, mean, ce1, asrc1, adst1,
                                           mx1, den1, xl1, agg1);
  finalize1_kernel<<<((N_NODES * F1) / 4 + 255) / 256, 256, 0, stream>>>(agg1, b1);

  // ----- layer 2 -----
  gemm2_wmma<<<N_NODES / 16, 64, 0, stream>>>(agg1, W2, xl2);
  alpha_node_kernel<H2, C2><<<(N_NODES * H2 + 255) / 256, 256, 0, stream>>>(
      xl2, as2, ad2, asrc2, adst2);
  const int g2 = (E_TOT * H2 + 255) / 256;
  edge_max_kernel<H2><<<g2, 256, 0, stream>>>(ei, ea, mean, ce2, asrc2, adst2, mx2);
  edge_den_kernel<H2><<<g2, 256, 0, stream>>>(ei, ea, mean, ce2, asrc2, adst2, mx2, den2);
  edge_agg2_kernel<<<g2, 256, 0, stream>>>(ei, ea, mean, ce2, asrc2, adst2,
                                           mx2, den2, xl2, agg2);
  final2_kernel<<<(N_NODES * C2 + 255) / 256, 256, 0, stream>>>(agg2, b2, out);
}